// EmbeddingHead_74775380624051
// MI455X (gfx1250) — compile-verified
//
#include <hip/hip_runtime.h>

typedef __attribute__((ext_vector_type(2))) float v2f;
typedef __attribute__((ext_vector_type(8))) float v8f;

#define BATCH 25
#define LSEQ  11
#define NOUT  4
#define DIM   2048
#define VOCAB 32000
#define KSEL  1024

#define TILE_N          16
#define WAVES_PER_BLOCK 8
#define BLOCK_THREADS   256
#define BLOCKS_PER_BL   (KSEL / (TILE_N * WAVES_PER_BLOCK))  // 1024/128 = 8

__global__ __launch_bounds__(BLOCK_THREADS)
void gather_head_wmma_f32(const float* __restrict__ h,
                          const float* __restrict__ w,
                          const int*   __restrict__ index,
                          float*       __restrict__ out)
{
    // rows 0..3 = h[b,l,o,:], row 4 = zeros (pads M=4 up to the 16x16 WMMA tile)
    __shared__ float hs[(NOUT + 1) * DIM];

    const int bl   = blockIdx.x / BLOCKS_PER_BL;   // which (b,l) pair
    const int nblk = blockIdx.x % BLOCKS_PER_BL;   // which 128-wide N slab
    const int tid  = threadIdx.x;
    const int lane = tid & 31;
    const int wave = tid >> 5;

    // Cooperative LDS fill: h[b,l] is 4x2048 f32 = 32KB (+8KB zero row)
    {
        const float4* src = (const float4*)(h + (size_t)bl * (NOUT * DIM));
        float4* dst = (float4*)hs;
        #pragma unroll
        for (int i = tid; i < (NOUT * DIM) / 4; i += BLOCK_THREADS)
            dst[i] = src[i];
        float4 z = {0.f, 0.f, 0.f, 0.f};
        float4* zr = (float4*)(hs + NOUT * DIM);
        #pragma unroll
        for (int i = tid; i < DIM / 4; i += BLOCK_THREADS)
            zr[i] = z;
    }
    __syncthreads();

    // This wave's 16-column output tile
    const int n0   = nblk * (TILE_N * WAVES_PER_BLOCK) + wave * TILE_N;
    const int nl   = lane & 15;              // N (and M) index this lane serves
    const int koff = (lane >= 16) ? 2 : 0;   // K sub-offset per f32 WMMA layout

    // Gathered vocab row for this lane's output column (computed once)
    const int row = index[(size_t)bl * KSEL + n0 + nl];
    const float* wp = w + (size_t)row * DIM + koff;

    // A-matrix source row in LDS (clamped to the zero row for M >= 4)
    const int am = (nl < NOUT) ? nl : NOUT;
    const float* ap = hs + am * DIM + koff;

    v8f acc0 = {};
    v8f acc1 = {};
    #pragma unroll 8
    for (int k = 0; k < DIM; k += 8) {
        v2f a0 = *(const v2f*)(ap + k);
        v2f b0 = *(const v2f*)(wp + k);
        v2f a1 = *(const v2f*)(ap + k + 4);
        v2f b1 = *(const v2f*)(wp + k + 4);
        acc0 = __builtin_amdgcn_wmma_f32_16x16x4_f32(false, a0, false, b0,
                                                     (short)0, acc0, false, false);
        acc1 = __builtin_amdgcn_wmma_f32_16x16x4_f32(false, a1, false, b1,
                                                     (short)0, acc1, false, false);
    }
    v8f acc = acc0 + acc1;

    // D layout: VGPR r, lanes 0-15 -> M=r (o=r), N = n0 + lane. Only o=0..3 needed.
    if (lane < 16) {
        float* ob = out + (size_t)bl * NOUT * KSEL + n0 + lane;
        #pragma unroll
        for (int r = 0; r < NOUT; ++r)
            ob[(size_t)r * KSEL] = acc[r];
    }
}

extern "C" void kernel_launch(void* const* d_in, const int* in_sizes, int n_in,
                              void* d_out, int out_size, void* d_ws, size_t ws_size,
                              hipStream_t stream)
{
    const float* h   = (const float*)d_in[0];
    const float* w   = (const float*)d_in[1];
    const int*   idx = (const int*)d_in[2];
    float* out = (float*)d_out;

    dim3 grid(BATCH * LSEQ * BLOCKS_PER_BL);  // 25*11*8 = 2200 blocks
    dim3 block(BLOCK_THREADS);                // 8 waves, one 16-col tile each
    hipLaunchKernelGGL(gather_head_wmma_f32, grid, block, 0, stream,
                       h, w, idx, out);
}